// ChebConvLayer_3908420239973
// MI455X (gfx1250) — compile-verified
//
#include <hip/hip_runtime.h>
#include <stdint.h>

// ---------------------------------------------------------------------------
// ChebConv forward for MI455X (gfx1250, wave32, WMMA).
//   Phase 1: degree -> deg^-1/2 -> per-edge Laplacian weight w_e
//   Phase 2: three scatter props with recurrence folded in:
//            T1 = prop(x);  T2 = 2*prop(T1) - x;  T3 = 2*prop(T2) - T1
//   Phase 3: out = [x|T1|T2|T3] (N x 192) @ W_flat (192 x 48) + bias
//            via V_WMMA_F32_16X16X4_F32 (full f32 precision).
//   W is staged in LDS pair-interleaved so every B operand is one aligned
//   ds_load_b64 with an immediate offset -> no register repacking movs.
// ---------------------------------------------------------------------------

typedef __attribute__((ext_vector_type(2))) float v2f;
typedef __attribute__((ext_vector_type(8))) float v8f;

#define FD   48          // feature dim (D_IN == D_OUT)
#define KORD 4           // Chebyshev order
#define KBIG (KORD * FD) // 192 = concatenated K dimension

// -------------------------- phase 1 kernels --------------------------------

__global__ __launch_bounds__(256) void cheb_deg_kernel(
    const long long* __restrict__ row, const long long* __restrict__ col,
    unsigned* __restrict__ deg, int n_edges)
{
  int e = blockIdx.x * blockDim.x + threadIdx.x;
  if (e < n_edges) {
    long long r = row[e];
    if (r != col[e]) atomicAdd(&deg[(int)r], 1u);
  }
}

__global__ __launch_bounds__(256) void cheb_dis_kernel(
    const unsigned* __restrict__ deg, float* __restrict__ dis, int n_nodes)
{
  int i = blockIdx.x * blockDim.x + threadIdx.x;
  if (i < n_nodes) {
    unsigned d = deg[i];
    dis[i] = (d > 0u) ? rsqrtf((float)d) : 0.0f;
  }
}

__global__ __launch_bounds__(256) void cheb_ew_kernel(
    const long long* __restrict__ row, const long long* __restrict__ col,
    const float* __restrict__ dis, float* __restrict__ w, int n_edges)
{
  int e = blockIdx.x * blockDim.x + threadIdx.x;
  if (e < n_edges) {
    long long r = row[e], c = col[e];
    w[e] = (r != c) ? (-dis[(int)r] * dis[(int)c]) : 0.0f;
  }
}

// -------------------------- phase 2 kernels --------------------------------

// dst = -src (initializes T_k accumulator with the -T_{k-2} recurrence term)
__global__ __launch_bounds__(256) void cheb_negcopy_kernel(
    const float* __restrict__ src, float* __restrict__ dst, int n)
{
  int i = blockIdx.x * blockDim.x + threadIdx.x;
  if (i < n) dst[i] = -src[i];
}

// acc[col[e]*48 + f] += scale * w[e] * h[row[e]*48 + f]
// One thread per (edge, feature). 48 consecutive threads share one edge,
// so edge metadata loads broadcast from cache; gathers/atomics stay in L2.
__global__ __launch_bounds__(256) void cheb_scatter_kernel(
    const long long* __restrict__ row, const long long* __restrict__ col,
    const float* __restrict__ w, const float* __restrict__ h,
    float* __restrict__ acc, float scale, int n_edges)
{
  int idx = blockIdx.x * blockDim.x + threadIdx.x;
  int total = n_edges * FD;                      // 76.8M fits in int32
  if (idx >= total) return;
  int e = idx / FD;
  int f = idx - e * FD;
  float we = w[e];
  if (we != 0.0f) {
    int s = (int)row[e];
    int d = (int)col[e];
    atomicAdd(acc + (size_t)d * FD + f, (we * scale) * h[(size_t)s * FD + f]);
  }
}

// -------------------------- phase 3: WMMA GEMM -----------------------------
// One wave -> 16x48 output slab (3 accumulators of v8f).
// A  (16x4 f32, 2 VGPRs): lane l<16: A[m0+l][k0..k0+1]; lanes 16-31: K+2..K+3.
// B  ( 4x16 f32, 2 VGPRs): lane = N column; VGPR0 = rows K0/K2, VGPR1 = K1/K3.
// C/D (16x16 f32, 8 VGPRs): VGPR r: lanes 0-15 -> M=r, lanes 16-31 -> M=r+8.
//
// LDS W layout (pair-interleaved): sW viewed as v2f[96][48];
//   sW2[kpair][n] = { W[2*kpair][n], W[2*kpair+1][n] }
// so a lane's B half-operand is one 8-byte aligned ds_load_b64, and all
// offsets are compile-time immediates off a single lane-dependent base.

__global__ __launch_bounds__(256) void cheb_gemm_kernel(
    const float* __restrict__ T0, const float* __restrict__ T1,
    const float* __restrict__ T2, const float* __restrict__ T3,
    const float* __restrict__ W,     // 192 x 48 row-major (== weight flat)
    const float* __restrict__ bias,  // 48
    float* __restrict__ out, int n_nodes)
{
  __shared__ float sW[KBIG * FD];    // 36 KB of 320 KB LDS
  for (int i = threadIdx.x; i < KBIG * FD; i += 256) {
    int k = i / FD;
    int n = i - k * FD;
    sW[(k >> 1) * (2 * FD) + n * 2 + (k & 1)] = W[i];
  }
  __syncthreads();

  const int lane = threadIdx.x & 31;
  const int wave = threadIdx.x >> 5;              // 8 waves per block
  const int m0   = (blockIdx.x * 8 + wave) * 16;  // 16-row slab base
  if (m0 + 16 > n_nodes) return;                  // whole-wave exit: EXEC stays all-1s

  const int nn  = lane & 15;        // N column within a 16-wide tile (== M row for A)
  const int kh  = (lane >> 4) << 1; // 0 (lanes 0-15) or 2 (lanes 16-31)
  const int khh = lane >> 4;        // kh / 2

  // Lane-dependent base into the pair-interleaved LDS image of W.
  const v2f* __restrict__ bbase = ((const v2f*)sW) + (khh * FD + nn);

  v8f acc0, acc1, acc2;
  {
    float b0 = bias[nn], b1 = bias[16 + nn], b2 = bias[32 + nn];
#pragma unroll
    for (int r = 0; r < 8; ++r) { acc0[r] = b0; acc1[r] = b1; acc2[r] = b2; }
  }

#pragma unroll
  for (int q = 0; q < KORD; ++q) {
    const float* Tq = (q == 0) ? T0 : (q == 1) ? T1 : (q == 2) ? T2 : T3;
    const float* arow = Tq + (size_t)(m0 + nn) * FD + kh;
#pragma unroll
    for (int kc = 0; kc < FD; kc += 4) {
      v2f a = *(const v2f*)(arow + kc);        // 8B aligned global_load_b64
      const int kp = ((q * FD + kc) >> 1) * FD;  // compile-time per iteration
      v2f b0 = bbase[kp +  0];                 // ds_load_b64, immediate offset
      v2f b1 = bbase[kp + 16];
      v2f b2 = bbase[kp + 32];
      acc0 = __builtin_amdgcn_wmma_f32_16x16x4_f32(false, a, false, b0,
                                                   (short)0, acc0, false, false);
      acc1 = __builtin_amdgcn_wmma_f32_16x16x4_f32(false, a, false, b1,
                                                   (short)0, acc1, false, false);
      acc2 = __builtin_amdgcn_wmma_f32_16x16x4_f32(false, a, false, b2,
                                                   (short)0, acc2, false, false);
    }
  }

  const int rbase = m0 + ((lane >> 4) << 3);  // lanes 16-31 hold rows M+8..M+15
#pragma unroll
  for (int r = 0; r < 8; ++r) {
    float* orow = out + (size_t)(rbase + r) * FD + nn;
    orow[0]  = acc0[r];
    orow[16] = acc1[r];
    orow[32] = acc2[r];
  }
}

// Scalar tail for M not divisible by 16 (never launched for N=50000).
__global__ __launch_bounds__(256) void cheb_gemm_tail_kernel(
    const float* __restrict__ T0, const float* __restrict__ T1,
    const float* __restrict__ T2, const float* __restrict__ T3,
    const float* __restrict__ W, const float* __restrict__ bias,
    float* __restrict__ out, int row_start, int n_nodes)
{
  int idx = blockIdx.x * blockDim.x + threadIdx.x;
  int m = row_start + idx / FD;
  int n = idx % FD;
  if (m >= n_nodes) return;
  const float* Ts[4] = {T0, T1, T2, T3};
  float s = bias[n];
  for (int q = 0; q < KORD; ++q) {
    const float* t = Ts[q] + (size_t)m * FD;
    for (int c = 0; c < FD; ++c) s += t[c] * W[(q * FD + c) * FD + n];
  }
  out[(size_t)m * FD + n] = s;
}

// ------------------------------- launcher ----------------------------------

extern "C" void kernel_launch(void* const* d_in, const int* in_sizes, int n_in,
                              void* d_out, int out_size, void* d_ws, size_t ws_size,
                              hipStream_t stream)
{
  const float*     x    = (const float*)d_in[0];
  const long long* ei   = (const long long*)d_in[1];   // int64, shape (2, E) flat
  const float*     W    = (const float*)d_in[2];       // (K, 48, 48) == 192x48 flat
  const float*     bias = (const float*)d_in[3];
  float*           out  = (float*)d_out;

  const int n_nodes = in_sizes[0] / FD;
  const int n_edges = in_sizes[1] / 2;
  const long long* row = ei;             // source j
  const long long* col = ei + n_edges;   // target i

  // Workspace carve-up (~35.6 MB for the given sizes)
  char* ws = (char*)d_ws;
  const size_t tbytes = (size_t)n_nodes * FD * sizeof(float);
  float*    T1  = (float*)ws;    ws += tbytes;
  float*    T2  = (float*)ws;    ws += tbytes;
  float*    T3  = (float*)ws;    ws += tbytes;
  float*    wE  = (float*)ws;    ws += (size_t)n_edges * sizeof(float);
  float*    dis = (float*)ws;    ws += (size_t)n_nodes * sizeof(float);
  unsigned* deg = (unsigned*)ws; ws += (size_t)n_nodes * sizeof(unsigned);
  (void)ws_size; (void)n_in; (void)out_size;

  const dim3 blk(256);
  const int eb = (n_edges + 255) / 256;
  const int nb = (n_nodes + 255) / 256;
  const int cb = (int)(((long long)n_nodes * FD + 255) / 256);
  const int sb = (int)(((long long)n_edges * FD + 255) / 256);

  // Phase 1: normalization weights
  hipMemsetAsync(deg, 0, (size_t)n_nodes * sizeof(unsigned), stream);
  cheb_deg_kernel<<<eb, blk, 0, stream>>>(row, col, deg, n_edges);
  cheb_dis_kernel<<<nb, blk, 0, stream>>>(deg, dis, n_nodes);
  cheb_ew_kernel <<<eb, blk, 0, stream>>>(row, col, dis, wE, n_edges);

  // Phase 2: Chebyshev recurrence via scatter (recurrence folded into init)
  hipMemsetAsync(T1, 0, tbytes, stream);
  cheb_scatter_kernel<<<sb, blk, 0, stream>>>(row, col, wE, x,  T1, 1.0f, n_edges);

  cheb_negcopy_kernel<<<cb, blk, 0, stream>>>(x,  T2, n_nodes * FD);
  cheb_scatter_kernel<<<sb, blk, 0, stream>>>(row, col, wE, T1, T2, 2.0f, n_edges);

  cheb_negcopy_kernel<<<cb, blk, 0, stream>>>(T1, T3, n_nodes * FD);
  cheb_scatter_kernel<<<sb, blk, 0, stream>>>(row, col, wE, T2, T3, 2.0f, n_edges);

  // Phase 3: fused GEMM  out = [x|T1|T2|T3] @ W_flat + bias  (WMMA f32)
  const int mtiles = n_nodes / 16;           // 3125 for N=50000 (exact)
  const int gb = (mtiles + 7) / 8;
  if (gb > 0)
    cheb_gemm_kernel<<<gb, blk, 0, stream>>>(x, T1, T2, T3, W, bias, out, n_nodes);

  const int tail_rows = n_nodes - mtiles * 16;  // 0 for N=50000
  if (tail_rows > 0) {
    const int tb = (tail_rows * FD + 255) / 256;
    cheb_gemm_tail_kernel<<<tb, blk, 0, stream>>>(x, T1, T2, T3, W, bias, out,
                                                  mtiles * 16, n_nodes);
  }
}